// MultiTaskComplexGNN_51943334478500
// MI455X (gfx1250) — compile-verified
//
#include <hip/hip_runtime.h>
#include <hip/hip_bf16.h>

typedef float v2f __attribute__((ext_vector_type(2)));
typedef float v8f __attribute__((ext_vector_type(8)));

#define WMMA_F32(a, b, c) \
    __builtin_amdgcn_wmma_f32_16x16x4_f32(false, (a), false, (b), (short)0, (c), false, false)

// ---------------------------------------------------------------------------
// Zero fill (grid-stride)
// ---------------------------------------------------------------------------
__global__ __launch_bounds__(256) void k_zero(float* __restrict__ p, long n) {
    long i = (long)blockIdx.x * blockDim.x + threadIdx.x;
    long stride = (long)gridDim.x * blockDim.x;
    for (; i < n; i += stride) p[i] = 0.0f;
}

// Pack W [K x 64] row-major into pair-interleaved LDS:
//   sWp[(k>>1)*128 + col*2 + (k&1)] = W[k*64 + col]
// so a B-fragment {W[2p][col], W[2p+1][col]} is one aligned ds_load_b64.
__device__ __forceinline__ void pack_weights(float* sWp, const float* W, int K, int tid) {
    for (int i = tid; i < K * 64; i += 256) {
        int k = i >> 6, col = i & 63;
        sWp[(k >> 1) * 128 + (col << 1) + (k & 1)] = W[i];
    }
}

// ---------------------------------------------------------------------------
// h = relu(x @ W_in + b_in)   x:[N,128]  W_in:[128,64]  h:[N,64]
// 256 threads = 8 wave32; each wave computes a 16x64 tile via WMMA f32 16x16x4.
// Out-of-range rows are address-clamped (loads stay unconditional: WMMA rows
// are independent, garbage rows are never stored).
// ---------------------------------------------------------------------------
__global__ __launch_bounds__(256) void k_gemm_in(const float* __restrict__ x,
                                                 const float* __restrict__ Win,
                                                 const float* __restrict__ bin,
                                                 float* __restrict__ h, int N) {
    __shared__ float sWp[128 * 64];   // pair-interleaved W_in
    const int tid = threadIdx.x;
    pack_weights(sWp, Win, 128, tid);
    __syncthreads();

    const int wave  = tid >> 5;
    const int lane  = tid & 31;
    const int lrow  = lane & 15;           // A row / B,C col within tile
    const int hi    = lane >> 4;           // half-wave select
    const int khalf = hi << 1;
    const int m0    = blockIdx.x * 128 + wave * 16;
    const int row   = m0 + lrow;
    const int srow  = (row < N) ? row : (N - 1);   // clamped, always in-bounds
    const float* xrow = x + (size_t)srow * 128;

    v8f acc0 = {}, acc1 = {}, acc2 = {}, acc3 = {};
    for (int k = 0; k < 128; k += 4) {
        v2f a = *(const v2f*)&xrow[k + khalf];
        const float* Bp = &sWp[((k >> 1) + hi) * 128];
        v2f b;
        b = *(const v2f*)&Bp[(lrow +  0) << 1]; acc0 = WMMA_F32(a, b, acc0);
        b = *(const v2f*)&Bp[(lrow + 16) << 1]; acc1 = WMMA_F32(a, b, acc1);
        b = *(const v2f*)&Bp[(lrow + 32) << 1]; acc2 = WMMA_F32(a, b, acc2);
        b = *(const v2f*)&Bp[(lrow + 48) << 1]; acc3 = WMMA_F32(a, b, acc3);
    }

    const int rbase = m0 + (hi << 3);
    const float bb0 = bin[lrow], bb1 = bin[lrow + 16], bb2 = bin[lrow + 32], bb3 = bin[lrow + 48];
    for (int i = 0; i < 8; ++i) {
        int r = rbase + i;
        if (r < N) {
            float* orow = h + (size_t)r * 64;
            orow[lrow +  0] = fmaxf(acc0[i] + bb0, 0.0f);
            orow[lrow + 16] = fmaxf(acc1[i] + bb1, 0.0f);
            orow[lrow + 32] = fmaxf(acc2[i] + bb2, 0.0f);
            orow[lrow + 48] = fmaxf(acc3[i] + bb3, 0.0f);
        }
    }
}

// ---------------------------------------------------------------------------
// Scatter-add: agg[dst[e]] += h[src[e]]  for all edges. 16 lanes per edge,
// each lane moves 4 consecutive floats (coalesced 256B gather per edge),
// HW f32 atomics on the L2-resident agg buffer.
// ---------------------------------------------------------------------------
__global__ __launch_bounds__(256) void k_scatter(const float* __restrict__ h,
                                                 const int* __restrict__ ei,
                                                 float* __restrict__ agg,
                                                 int E, int N) {
    int t   = blockIdx.x * 256 + threadIdx.x;
    int e   = t >> 4;
    int sub = (t & 15) << 2;
    if (e >= E) return;
    int s = ei[e];
    int d = ei[E + e];
    if ((unsigned)s >= (unsigned)N || (unsigned)d >= (unsigned)N) return;
    const float4 v = *(const float4*)(h + (size_t)s * 64 + sub);
    float* ap = agg + (size_t)d * 64 + sub;
    unsafeAtomicAdd(ap + 0, v.x);
    unsafeAtomicAdd(ap + 1, v.y);
    unsafeAtomicAdd(ap + 2, v.z);
    unsafeAtomicAdd(ap + 3, v.w);
}

// ---------------------------------------------------------------------------
// Fused GIN MLP: h = relu(relu((h+agg) @ W1 + b1) @ W2 + b2), in place.
// Intermediate t staged through LDS to re-swizzle C-layout -> A-layout.
// ---------------------------------------------------------------------------
__global__ __launch_bounds__(256) void k_gin_mlp(float* __restrict__ h,
                                                 const float* __restrict__ agg,
                                                 const float* __restrict__ W1,
                                                 const float* __restrict__ b1,
                                                 const float* __restrict__ W2,
                                                 const float* __restrict__ b2,
                                                 int N) {
    __shared__ float sW1p[64 * 64];
    __shared__ float sW2p[64 * 64];
    __shared__ float sT[8 * 16 * 64];   // per-wave 16x64 intermediate tiles
    const int tid = threadIdx.x;
    pack_weights(sW1p, W1, 64, tid);
    pack_weights(sW2p, W2, 64, tid);
    __syncthreads();

    const int wave  = tid >> 5;
    const int lane  = tid & 31;
    const int lrow  = lane & 15;
    const int hi    = lane >> 4;
    const int khalf = hi << 1;
    const int m0    = blockIdx.x * 128 + wave * 16;
    const int row   = m0 + lrow;
    const int srow  = (row < N) ? row : (N - 1);
    const float* hrow = h   + (size_t)srow * 64;
    const float* grow = agg + (size_t)srow * 64;

    // Stage 1: acc = (h + agg) @ W1
    v8f acc0 = {}, acc1 = {}, acc2 = {}, acc3 = {};
    for (int k = 0; k < 64; k += 4) {
        v2f ah = *(const v2f*)&hrow[k + khalf];
        v2f ag = *(const v2f*)&grow[k + khalf];
        v2f a  = ah + ag;
        const float* Bp = &sW1p[((k >> 1) + hi) * 128];
        v2f b;
        b = *(const v2f*)&Bp[(lrow +  0) << 1]; acc0 = WMMA_F32(a, b, acc0);
        b = *(const v2f*)&Bp[(lrow + 16) << 1]; acc1 = WMMA_F32(a, b, acc1);
        b = *(const v2f*)&Bp[(lrow + 32) << 1]; acc2 = WMMA_F32(a, b, acc2);
        b = *(const v2f*)&Bp[(lrow + 48) << 1]; acc3 = WMMA_F32(a, b, acc3);
    }

    // t = relu(acc + b1) -> LDS (wave-private 16x64 tile)
    {
        float* tw = &sT[wave * 16 * 64];
        const float bb0 = b1[lrow], bb1 = b1[lrow + 16], bb2 = b1[lrow + 32], bb3 = b1[lrow + 48];
        const int rb = hi << 3;
        for (int i = 0; i < 8; ++i) {
            float* trow = tw + (rb + i) * 64;
            trow[lrow +  0] = fmaxf(acc0[i] + bb0, 0.0f);
            trow[lrow + 16] = fmaxf(acc1[i] + bb1, 0.0f);
            trow[lrow + 32] = fmaxf(acc2[i] + bb2, 0.0f);
            trow[lrow + 48] = fmaxf(acc3[i] + bb3, 0.0f);
        }
    }
    __syncthreads();

    // Stage 2: d = t @ W2
    v8f d0 = {}, d1 = {}, d2 = {}, d3 = {};
    {
        const float* tw = &sT[wave * 16 * 64] + lrow * 64;
        for (int k = 0; k < 64; k += 4) {
            v2f a = *(const v2f*)&tw[k + khalf];
            const float* Bp = &sW2p[((k >> 1) + hi) * 128];
            v2f b;
            b = *(const v2f*)&Bp[(lrow +  0) << 1]; d0 = WMMA_F32(a, b, d0);
            b = *(const v2f*)&Bp[(lrow + 16) << 1]; d1 = WMMA_F32(a, b, d1);
            b = *(const v2f*)&Bp[(lrow + 32) << 1]; d2 = WMMA_F32(a, b, d2);
            b = *(const v2f*)&Bp[(lrow + 48) << 1]; d3 = WMMA_F32(a, b, d3);
        }
    }

    const float cb0 = b2[lrow], cb1 = b2[lrow + 16], cb2 = b2[lrow + 32], cb3 = b2[lrow + 48];
    const int rbase = m0 + (hi << 3);
    for (int i = 0; i < 8; ++i) {
        int r = rbase + i;
        if (r < N) {
            float* orow = h + (size_t)r * 64;   // in-place: row-local dependency only
            orow[lrow +  0] = fmaxf(d0[i] + cb0, 0.0f);
            orow[lrow + 16] = fmaxf(d1[i] + cb1, 0.0f);
            orow[lrow + 32] = fmaxf(d2[i] + cb2, 0.0f);
            orow[lrow + 48] = fmaxf(d3[i] + cb3, 0.0f);
        }
    }
}

// ---------------------------------------------------------------------------
// Mean-pool accumulation exploiting sorted batch: each (quarter, col) lane
// walks 16 consecutive nodes, run-length accumulates per graph, flushes one
// atomic per run. Block covers 64 nodes x 64 cols.
// ---------------------------------------------------------------------------
__global__ __launch_bounds__(256) void k_pool(const float* __restrict__ h,
                                              const int* __restrict__ batch,
                                              float* __restrict__ sums,
                                              float* __restrict__ counts,
                                              int N, int G) {
    const int c = threadIdx.x & 63;
    const int q = threadIdx.x >> 6;                 // 0..3
    const long nb = (long)blockIdx.x * 64 + q * 16; // first node for this lane group

    float acc = 0.0f;
    int curg = -1;
    float cn = 0.0f;
    for (int i = 0; i < 16; ++i) {
        long n = nb + i;
        if (n >= N) break;
        int g = batch[n];
        if ((unsigned)g >= (unsigned)G) continue;
        if (g != curg) {
            if (curg >= 0) {
                unsafeAtomicAdd(&sums[(size_t)curg * 64 + c], acc);
                if (c == 0) unsafeAtomicAdd(&counts[curg], cn);
            }
            acc = 0.0f; cn = 0.0f; curg = g;
        }
        acc += h[(size_t)n * 64 + c];
        cn  += 1.0f;
    }
    if (curg >= 0) {
        unsafeAtomicAdd(&sums[(size_t)curg * 64 + c], acc);
        if (c == 0) unsafeAtomicAdd(&counts[curg], cn);
    }
}

// ---------------------------------------------------------------------------
// Finalize: h_graph = sums / max(count,1); pred_exp = hg@W_exp+b; pred_aux.
// out layout: [0,G*64) h_graph | [G*64,G*65) pred_exp | [G*65,G*69) pred_aux
// ---------------------------------------------------------------------------
__global__ __launch_bounds__(64) void k_finalize(float* __restrict__ out,
                                                 const float* __restrict__ counts,
                                                 const float* __restrict__ Wexp,
                                                 const float* __restrict__ bexp,
                                                 const float* __restrict__ Waux,
                                                 const float* __restrict__ baux,
                                                 int G) {
    __shared__ float hg[64];
    const int g = blockIdx.x;
    const int c = threadIdx.x;
    float cnt = fmaxf(counts[g], 1.0f);
    float v = out[(size_t)g * 64 + c] / cnt;
    out[(size_t)g * 64 + c] = v;
    hg[c] = v;
    __syncthreads();
    if (c == 0) {
        float acc = bexp[0];
        for (int i = 0; i < 64; ++i) acc += hg[i] * Wexp[i];
        out[(size_t)G * 64 + g] = acc;
    } else if (c <= 4) {
        int k = c - 1;
        float acc = baux[k];
        for (int i = 0; i < 64; ++i) acc += hg[i] * Waux[i * 4 + k];
        out[(size_t)G * 65 + (size_t)g * 4 + k] = acc;
    }
}

// ---------------------------------------------------------------------------
extern "C" void kernel_launch(void* const* d_in, const int* in_sizes, int n_in,
                              void* d_out, int out_size, void* d_ws, size_t ws_size,
                              hipStream_t stream) {
    const float* x     = (const float*)d_in[0];
    const int*   ei    = (const int*)d_in[1];
    const int*   batch = (const int*)d_in[2];
    // d_in[3] = num_graphs scalar (device); G derived on host instead.
    const float* Win  = (const float*)d_in[4];
    const float* bin  = (const float*)d_in[5];
    const float* W1_0 = (const float*)d_in[6];
    const float* b1_0 = (const float*)d_in[7];
    const float* W2_0 = (const float*)d_in[8];
    const float* b2_0 = (const float*)d_in[9];
    const float* W1_1 = (const float*)d_in[10];
    const float* b1_1 = (const float*)d_in[11];
    const float* W2_1 = (const float*)d_in[12];
    const float* b2_1 = (const float*)d_in[13];
    const float* Wexp = (const float*)d_in[14];
    const float* bexp = (const float*)d_in[15];
    const float* Waux = (const float*)d_in[16];
    const float* baux = (const float*)d_in[17];

    const int N = in_sizes[2];          // batch has one entry per node
    const int E = in_sizes[1] / 2;      // edge_index is [2, E]
    const int G = out_size / 69;        // 64 (h_graph) + 1 (exp) + 4 (aux)

    float* h      = (float*)d_ws;
    float* agg    = h + (size_t)N * 64;
    float* counts = agg + (size_t)N * 64;
    float* out    = (float*)d_out;

    const int gemmBlocks    = (N + 127) / 128;
    const int scatterBlocks = (E + 15) / 16;
    const int poolBlocks    = (N + 63) / 64;

    // Input layer
    k_gemm_in<<<gemmBlocks, 256, 0, stream>>>(x, Win, bin, h, N);

    // GIN layer 0
    k_zero<<<2048, 256, 0, stream>>>(agg, (long)N * 64);
    k_scatter<<<scatterBlocks, 256, 0, stream>>>(h, ei, agg, E, N);
    k_gin_mlp<<<gemmBlocks, 256, 0, stream>>>(h, agg, W1_0, b1_0, W2_0, b2_0, N);

    // GIN layer 1
    k_zero<<<2048, 256, 0, stream>>>(agg, (long)N * 64);
    k_scatter<<<scatterBlocks, 256, 0, stream>>>(h, ei, agg, E, N);
    k_gin_mlp<<<gemmBlocks, 256, 0, stream>>>(h, agg, W1_1, b1_1, W2_1, b2_1, N);

    // Pool + heads (accumulators zeroed every call for determinism)
    k_zero<<<64, 256, 0, stream>>>(out, (long)out_size);
    k_zero<<<1, 256, 0, stream>>>(counts, (long)G);
    k_pool<<<poolBlocks, 256, 0, stream>>>(h, batch, out, counts, N, G);
    k_finalize<<<G, 64, 0, stream>>>(out, counts, Wexp, bexp, Waux, baux, G);
}